// SSIMLoss_78297253806614
// MI455X (gfx1250) — compile-verified
//
#include <hip/hip_runtime.h>

typedef float v2f __attribute__((ext_vector_type(2)));
typedef float v4f __attribute__((ext_vector_type(4)));
typedef float v8f __attribute__((ext_vector_type(8)));

#define HW         512
#define OUTD       506          // 512 - 7 + 1
#define PLANE      (HW * HW)
#define NIMG       32
#define NPLANES    96           // 32 * 3
#define TILES_X    32           // ceil(506/16) -> 32 (last tile masked)
#define TILES_PP   (TILES_X * TILES_X)      // 1024 tiles per plane
#define NTILES     (TILES_PP * NPLANES)     // 98304
#define WPB        8                         // waves (tiles) per block
#define NBLK       (NTILES / WPB)            // 12288
#define TIN        22           // input tile extent (16 + 6)
#define NPAIR      11           // 22 rows as 11 even/odd row pairs (v2f)
#define TPADC      24           // padded LDS column count (v2f elements)
#define INV_NP     (1.0f / 49.0f)
#define COV_NORM   (49.0f / 48.0f)
#define TOTAL_PIX  24579456.0f  // 96 * 506 * 506

// ---------------- ws init: min/max slots (ws is poisoned with 0xAA) ---------
__global__ void ssim_init_kernel(unsigned* maxb, unsigned* minb) {
    int t = threadIdx.x;
    if (t < NIMG) {
        maxb[t] = 0u;            // all img2 values are >= 0
        minb[t] = 0x7F800000u;   // +inf bits
    }
}

// ---------------- per-image data_range: min/max of img2 ---------------------
// Non-negative floats order identically as unsigned bit patterns, so
// atomicMax/atomicMin on bits is exact and order-independent (deterministic).
// float4 loads keep this pass at the b128 bandwidth path.
__global__ void ssim_range_kernel(const float* __restrict__ img2,
                                  unsigned* __restrict__ maxb,
                                  unsigned* __restrict__ minb) {
    const int b = blockIdx.y;
    const v4f* p = (const v4f*)(img2 + (size_t)b * (3 * PLANE));
    const int tid = blockIdx.x * blockDim.x + threadIdx.x;   // 64*256 = 16384
    unsigned mx = 0u, mn = 0x7F800000u;
    for (int i = tid; i < (3 * PLANE) / 4; i += 64 * 256) {
        v4f v = p[i];
#pragma unroll
        for (int c = 0; c < 4; ++c) {
            unsigned u = __float_as_uint(v[c]);
            mx = mx > u ? mx : u;
            mn = mn < u ? mn : u;
        }
    }
#pragma unroll
    for (int off = 16; off > 0; off >>= 1) {
        unsigned omx = __shfl_xor(mx, off, 32);
        unsigned omn = __shfl_xor(mn, off, 32);
        mx = mx > omx ? mx : omx;
        mn = mn < omn ? mn : omn;
    }
    if ((threadIdx.x & 31) == 0) {
        atomicMax(&maxb[b], mx);
        atomicMin(&minb[b], mn);
    }
}

// ---------------- fused box-filter (WMMA) + SSIM map + reduction ------------
// One wave per 16x16 output tile. Vertical box pass as S = A(16x22 band-ones)
// x H(22x16) via chained V_WMMA_F32_16X16X4_F32 (K padded 22->24). Horizontal
// 7-sums H are built per-lane with packed-f32 math: LDS tiles are stored
// row-pair interleaved (v2f = {row 2p, row 2p+1}) so one ds_load_b64 feeds
// both K-components a lane owns per WMMA step, and the five running sums are
// v_pk_add/v_pk_fma — the result v2f IS the B fragment (j=0 -> .x, j=1 -> .y).
__global__ void ssim_map_kernel(const float* __restrict__ img1,
                                const float* __restrict__ img2,
                                const unsigned* __restrict__ maxb,
                                const unsigned* __restrict__ minb,
                                float* __restrict__ partial) {
    __shared__ v2f ldsP[WPB][2][NPAIR][TPADC];   // 8*2*11*24*8B = 33792 B
    __shared__ float wsum[WPB];

    const int lane = threadIdx.x & 31;
    const int wave = threadIdx.x >> 5;
    const int tile = blockIdx.x * WPB + wave;         // exact cover: no guard
    const int plane = tile >> 10;                     // / TILES_PP
    const int t     = tile & (TILES_PP - 1);
    const int r0 = (t >> 5) << 4;
    const int c0 = (t & 31) << 4;
    const size_t pbase = (size_t)plane * PLANE;

    // ---- stage 22x22 (clamped) tiles, row-pair interleaved, both images ----
    for (int idx = lane; idx < TIN * TIN; idx += 32) {
        const int rr = idx / TIN;
        const int cc = idx - rr * TIN;
        const int r = r0 + rr, c = c0 + cc;
        float xv = 0.0f, yv = 0.0f;
        if (r < HW && c < HW) {
            const size_t o = pbase + (size_t)r * HW + c;
            xv = img1[o];
            yv = img2[o];
        }
        ((float*)&ldsP[wave][0][rr >> 1][cc])[rr & 1] = xv;
        ((float*)&ldsP[wave][1][rr >> 1][cc])[rr & 1] = yv;
    }
    __syncthreads();

    const int n  = lane & 15;      // output column / A-matrix row M
    const int hi = lane >> 4;      // lane half selects K sub-pair

    v8f accX = {}, accY = {}, accXX = {}, accYY = {}, accXY = {};

#pragma unroll
    for (int s = 0; s < 6; ++s) {
        const int p = 2 * s + hi;              // row pair this lane sums
        v2f sx = {0.f, 0.f}, sy = {0.f, 0.f};
        v2f sxx = {0.f, 0.f}, syy = {0.f, 0.f}, sxy = {0.f, 0.f};
        if (p < NPAIR) {                       // only (s=5,hi=1) is padding
            const v2f* rx = &ldsP[wave][0][p][n];
            const v2f* ry = &ldsP[wave][1][p][n];
#pragma unroll
            for (int tt = 0; tt < 7; ++tt) {
                const v2f x = rx[tt];
                const v2f y = ry[tt];
                sx += x;            sy += y;
                sxx += x * x;       syy += y * y;       sxy += x * y;
            }
        }
        // banded ones A[m][k] = 1 iff m <= k <= m+6 (0 in K-pad region)
        const int k0 = 4 * s + 2 * hi;
        v2f a;
        a.x = (k0     >= n && k0     <= n + 6) ? 1.0f : 0.0f;
        a.y = (k0 + 1 >= n && k0 + 1 <= n + 6) ? 1.0f : 0.0f;

        accX  = __builtin_amdgcn_wmma_f32_16x16x4_f32(false, a, false, sx,  (short)0, accX,  false, false);
        accY  = __builtin_amdgcn_wmma_f32_16x16x4_f32(false, a, false, sy,  (short)0, accY,  false, false);
        accXX = __builtin_amdgcn_wmma_f32_16x16x4_f32(false, a, false, sxx, (short)0, accXX, false, false);
        accYY = __builtin_amdgcn_wmma_f32_16x16x4_f32(false, a, false, syy, (short)0, accYY, false, false);
        accXY = __builtin_amdgcn_wmma_f32_16x16x4_f32(false, a, false, sxy, (short)0, accXY, false, false);
    }

    // ---- per-image constants ----
    const int b = plane / 3;
    const float dr = __uint_as_float(maxb[b]) - __uint_as_float(minb[b]);
    const float C1 = (0.01f * dr) * (0.01f * dr);
    const float C2 = (0.03f * dr) * (0.03f * dr);

    // ---- packed elementwise SSIM map on D-fragment (M = j + 8*hi, N = n) ---
    float lsum = 0.0f;
    const bool cok = (c0 + n) < OUTD;
#pragma unroll
    for (int t2 = 0; t2 < 4; ++t2) {
        const int j0 = 2 * t2;
        v2f Sx  = {accX[j0],  accX[j0 + 1]};
        v2f Sy  = {accY[j0],  accY[j0 + 1]};
        v2f Sxx = {accXX[j0], accXX[j0 + 1]};
        v2f Syy = {accYY[j0], accYY[j0 + 1]};
        v2f Sxy = {accXY[j0], accXY[j0 + 1]};
        v2f ux  = Sx  * INV_NP;
        v2f uy  = Sy  * INV_NP;
        v2f uxx = Sxx * INV_NP;
        v2f uyy = Syy * INV_NP;
        v2f uxy = Sxy * INV_NP;
        v2f vx  = (uxx - ux * ux) * COV_NORM;
        v2f vy  = (uyy - uy * uy) * COV_NORM;
        v2f vxy = (uxy - ux * uy) * COV_NORM;
        v2f num = (ux * uy * 2.0f + C1) * (vxy * 2.0f + C2);
        v2f den = (ux * ux + uy * uy + C1) * (vx + vy + C2);
        v2f S = num / den;
        const int m0 = j0 + 8 * hi;
        if (cok && (r0 + m0)     < OUTD) lsum += S.x;
        if (cok && (r0 + m0 + 1) < OUTD) lsum += S.y;
    }

    // ---- deterministic wave + block reduction ----
#pragma unroll
    for (int off = 16; off > 0; off >>= 1) lsum += __shfl_down(lsum, off, 32);
    if (lane == 0) wsum[wave] = lsum;
    __syncthreads();
    if (threadIdx.x == 0) {
        float sb = 0.0f;
#pragma unroll
        for (int w = 0; w < WPB; ++w) sb += wsum[w];
        partial[blockIdx.x] = sb;
    }
}

// ---------------- final fixed-order reduction -------------------------------
__global__ void ssim_final_kernel(const float* __restrict__ partial,
                                  float* __restrict__ out) {
    __shared__ float sh[256];
    float s = 0.0f;
    for (int i = threadIdx.x; i < NBLK; i += 256) s += partial[i];
    sh[threadIdx.x] = s;
    __syncthreads();
    for (int st = 128; st > 0; st >>= 1) {
        if (threadIdx.x < st) sh[threadIdx.x] += sh[threadIdx.x + st];
        __syncthreads();
    }
    if (threadIdx.x == 0) out[0] = 1.0f - sh[0] / TOTAL_PIX;
}

extern "C" void kernel_launch(void* const* d_in, const int* in_sizes, int n_in,
                              void* d_out, int out_size, void* d_ws, size_t ws_size,
                              hipStream_t stream) {
    (void)in_sizes; (void)n_in; (void)out_size; (void)ws_size;
    const float* img1 = (const float*)d_in[0];
    const float* img2 = (const float*)d_in[1];
    float* out = (float*)d_out;

    unsigned* maxb    = (unsigned*)d_ws;
    unsigned* minb    = maxb + NIMG;
    float*    partial = (float*)((char*)d_ws + 256);

    ssim_init_kernel<<<1, 32, 0, stream>>>(maxb, minb);
    ssim_range_kernel<<<dim3(64, NIMG), 256, 0, stream>>>(img2, maxb, minb);
    ssim_map_kernel<<<NBLK, 256, 0, stream>>>(img1, img2, maxb, minb, partial);
    ssim_final_kernel<<<1, 256, 0, stream>>>(partial, out);
}